// SpatialTemporal_80736795230316
// MI455X (gfx1250) — compile-verified
//
#include <hip/hip_runtime.h>
#include <cmath>

typedef __attribute__((ext_vector_type(2))) float v2f;
typedef __attribute__((ext_vector_type(8))) float v8f;

#define WMMA_F32(a, b, c) \
  __builtin_amdgcn_wmma_f32_16x16x4_f32(false, (a), false, (b), (short)0, (c), false, false)

namespace {
constexpr int kN    = 2000;   // nodes
constexpr int kB    = 4;      // batch
constexpr int kTin  = 34;     // input time
constexpr int kTout = 32;     // output time (dilation-2, kernel 2)
constexpr int kBT   = kB * kTout;        // 128 time-batches
constexpr long kRows = (long)kBT * kN;   // 256000 GEMM rows
constexpr int kTilesPerBT = kN / 16;     // 125
constexpr int kTiles = (int)(kRows / 16);// 16000
constexpr int kWaves = 8;                // waves per block
constexpr int kAS = 68;                  // padded LDS stride for 16x64 A stage
constexpr int kHS = 33;                  // padded LDS stride for 16x32 h stage
}

// ---------------------------------------------------------------------------
// Fused gated temporal conv (f32 WMMA, K=64 -> 64 outputs) + activation +
// GCN feature GEMM (f32 WMMA, K=32 -> 32 outputs). One 16-row tile per wave.
// ---------------------------------------------------------------------------
__global__ __launch_bounds__(256) void k_gate_gcn_wmma(
    const float* __restrict__ x,
    const float* __restrict__ g1w, const float* __restrict__ g1b,
    const float* __restrict__ g2w, const float* __restrict__ g2b,
    const float* __restrict__ gcw,
    float* __restrict__ hw)
{
  __shared__ float sWg[64 * 64];            // combined gate weights [k][j]
  __shared__ float sWc[32 * 32];            // gcn_w [d][e]
  __shared__ float sBias[64];               // gate1_b || gate2_b
  __shared__ __align__(16) float sA[kWaves][16 * kAS]; // per-wave stage

  const int tid = threadIdx.x;

  // Build combined gate weight: k = tap*32 + c, j<32 -> gate1 dc=j, else gate2.
  // gate*_w layout is (DC, C_IN, 1, 2) -> flat dc*64 + c*2 + tap.
  for (int i = tid; i < 64 * 64; i += 256) {
    int k = i >> 6, j = i & 63;
    int c = k & 31, tap = k >> 5;
    sWg[i] = (j < 32) ? g1w[j * 64 + c * 2 + tap]
                      : g2w[(j - 32) * 64 + c * 2 + tap];
  }
  for (int i = tid; i < 32 * 32; i += 256) sWc[i] = gcw[i];
  if (tid < 64) sBias[tid] = (tid < 32) ? g1b[tid] : g2b[tid - 32];
  __syncthreads();

  const int wv   = tid >> 5;
  const int lane = tid & 31;
  const int tile = blockIdx.x * kWaves + wv;        // exact: grid*8 == kTiles
  const int bt   = tile / kTilesPerBT;
  const int n0   = (tile - bt * kTilesPerBT) * 16;
  const int b    = bt >> 5;                          // bt / 32
  const int t    = bt & 31;                          // bt % 32

  // Stage A tile (16 rows x 64 K) into LDS with float4 loads.
  // Row m: K 0..31 = x[b,t,n0+m,:], K 32..63 = x[b,t+2,n0+m,:].
  const float4* __restrict__ x4 = reinterpret_cast<const float4*>(x);
  const long base0 = (long)(b * kTin + t)     * kN * 8;
  const long base2 = (long)(b * kTin + t + 2) * kN * 8;
  #pragma unroll
  for (int it = 0; it < 8; ++it) {
    int fidx = it * 32 + lane;          // 0..255 float4 slots
    int m = fidx >> 4, q = fidx & 15;
    float4 v = (q < 8) ? x4[base0 + (long)(n0 + m) * 8 + q]
                       : x4[base2 + (long)(n0 + m) * 8 + (q - 8)];
    *reinterpret_cast<float4*>(&sA[wv][m * kAS + q * 4]) = v;
  }
  __syncthreads();

  // WMMA f32 16x16x4 operand positions for this lane.
  const int col  = lane & 15;           // N index (and A row index)
  const int koff = (lane >> 4) << 1;    // 0 for lanes 0-15, 2 for 16-31
  const int mhi  = (lane >> 4) << 3;    // C/D row offset: 0 or 8

  // Gate GEMM: 16 K-steps, 4 column blocks (64 outputs), bias in C.
  const float b0 = sBias[col],      b1 = sBias[16 + col];
  const float b2 = sBias[32 + col], b3 = sBias[48 + col];
  v8f acc0 = {b0,b0,b0,b0,b0,b0,b0,b0};
  v8f acc1 = {b1,b1,b1,b1,b1,b1,b1,b1};
  v8f acc2 = {b2,b2,b2,b2,b2,b2,b2,b2};
  v8f acc3 = {b3,b3,b3,b3,b3,b3,b3,b3};

  const float* __restrict__ Arow = &sA[wv][col * kAS];
  #pragma unroll
  for (int kb = 0; kb < 16; ++kb) {
    const int k = kb * 4 + koff;
    v2f a; a.x = Arow[k]; a.y = Arow[k + 1];
    v2f w;
    w.x = sWg[k*64 +      col]; w.y = sWg[(k+1)*64 +      col]; acc0 = WMMA_F32(a, w, acc0);
    w.x = sWg[k*64 + 16 + col]; w.y = sWg[(k+1)*64 + 16 + col]; acc1 = WMMA_F32(a, w, acc1);
    w.x = sWg[k*64 + 32 + col]; w.y = sWg[(k+1)*64 + 32 + col]; acc2 = WMMA_F32(a, w, acc2);
    w.x = sWg[k*64 + 48 + col]; w.y = sWg[(k+1)*64 + 48 + col]; acc3 = WMMA_F32(a, w, acc3);
  }

  // h = tanh(gate1) * sigmoid(gate2); restage D-layout -> A-layout via LDS.
  #pragma unroll
  for (int i = 0; i < 8; ++i) {
    float h0 = tanhf(acc0[i]) * (1.f / (1.f + __expf(-acc2[i])));
    float h1 = tanhf(acc1[i]) * (1.f / (1.f + __expf(-acc3[i])));
    sA[wv][(i + mhi) * kHS +      col] = h0;
    sA[wv][(i + mhi) * kHS + 16 + col] = h1;
  }
  __syncthreads();

  // GCN GEMM: hw = h @ gcn_w, 8 K-steps, 2 column blocks.
  v8f o0 = {0,0,0,0,0,0,0,0}, o1 = {0,0,0,0,0,0,0,0};
  const float* __restrict__ Hrow = &sA[wv][col * kHS];
  #pragma unroll
  for (int kb = 0; kb < 8; ++kb) {
    const int k = kb * 4 + koff;
    v2f a; a.x = Hrow[k]; a.y = Hrow[k + 1];
    v2f w;
    w.x = sWc[k*32 +      col]; w.y = sWc[(k+1)*32 +      col]; o0 = WMMA_F32(a, w, o0);
    w.x = sWc[k*32 + 16 + col]; w.y = sWc[(k+1)*32 + 16 + col]; o1 = WMMA_F32(a, w, o1);
  }

  float* __restrict__ dst = hw + (long)tile * (16 * 32);
  #pragma unroll
  for (int i = 0; i < 8; ++i) {
    dst[(i + mhi) * 32 +      col] = o0[i];
    dst[(i + mhi) * 32 + 16 + col] = o1[i];
  }
}

// ---------------------------------------------------------------------------
// Graph-norm helpers + scatter
// ---------------------------------------------------------------------------
__global__ void k_deg_init(float* dis) {
  int n = blockIdx.x * 256 + threadIdx.x;
  if (n < kN) dis[n] = 1.0f;                       // self-loop weight
}

__global__ void k_deg_accum(const int* __restrict__ ei,
                            const float* __restrict__ ew, float* dis, int E) {
  int e = blockIdx.x * 256 + threadIdx.x;
  if (e < E) atomicAdd(&dis[ei[E + e]], ew[e]);    // segment_sum over col
}

__global__ void k_deg_finalize(float* dis) {
  int n = blockIdx.x * 256 + threadIdx.x;
  if (n < kN) { float d = dis[n]; dis[n] = d > 0.f ? rsqrtf(d) : 0.f; }
}

__global__ void k_edge_norm(const int* __restrict__ ei,
                            const float* __restrict__ ew,
                            const float* __restrict__ dis,
                            float* __restrict__ nrm, int E) {
  int e = blockIdx.x * 256 + threadIdx.x;
  if (e < E) nrm[e] = dis[ei[e]] * ew[e] * dis[ei[E + e]];
}

// out = gcn_b + hw * dis[n]^2   (bias + self-loop message, written once)
__global__ void k_out_init(const float* __restrict__ hw,
                           const float* __restrict__ dis,
                           const float* __restrict__ gcb,
                           float* __restrict__ out) {
  long i = (long)blockIdx.x * 256 + threadIdx.x;   // grid covers exactly
  int c = (int)(i & 31);
  int n = (int)((i >> 5) % kN);
  float d = dis[n];
  out[i] = gcb[c] + hw[i] * d * d;
}

// Edge scatter: 32 lanes = 32 channels of one edge -> coalesced atomics.
__global__ void k_scatter(const int* __restrict__ ei,
                          const float* __restrict__ nrm,
                          const float* __restrict__ hw,
                          float* __restrict__ out, int E) {
  int e = blockIdx.x * 8 + (threadIdx.x >> 5);
  if (e >= E) return;
  int c  = threadIdx.x & 31;
  int r  = ei[e];
  int cl = ei[E + e];
  float nm = nrm[e];
  long base = (long)blockIdx.y * ((long)kN * 32);
  float v = hw[base + (long)r * 32 + c] * nm;
  atomicAdd(&out[base + (long)cl * 32 + c], v);
}

// ---------------------------------------------------------------------------
extern "C" void kernel_launch(void* const* d_in, const int* in_sizes, int n_in,
                              void* d_out, int out_size, void* d_ws, size_t ws_size,
                              hipStream_t stream) {
  const float* x   = (const float*)d_in[0];
  const int*   ei  = (const int*)d_in[1];
  const float* ew  = (const float*)d_in[2];
  const float* g1w = (const float*)d_in[3];
  const float* g1b = (const float*)d_in[4];
  const float* g2w = (const float*)d_in[5];
  const float* g2b = (const float*)d_in[6];
  const float* gcw = (const float*)d_in[7];
  const float* gcb = (const float*)d_in[8];
  float* out = (float*)d_out;
  const int E = in_sizes[1] / 2;

  // Workspace layout: hw (256000*32 f32) | dis (2048 f32) | edge norms (E f32)
  float* hw  = (float*)d_ws;
  float* dis = hw + kRows * 32;
  float* nrm = dis + 2048;

  k_deg_init    <<<(kN + 255) / 256, 256, 0, stream>>>(dis);
  k_deg_accum   <<<(E + 255) / 256, 256, 0, stream>>>(ei, ew, dis, E);
  k_deg_finalize<<<(kN + 255) / 256, 256, 0, stream>>>(dis);
  k_edge_norm   <<<(E + 255) / 256, 256, 0, stream>>>(ei, ew, dis, nrm, E);

  k_gate_gcn_wmma<<<kTiles / kWaves, 256, 0, stream>>>(x, g1w, g1b, g2w, g2b,
                                                       gcw, hw);

  k_out_init<<<(int)((kRows * 32) / 256), 256, 0, stream>>>(hw, dis, gcb, out);

  dim3 sg((E + 7) / 8, kBT);
  k_scatter<<<sg, 256, 0, stream>>>(ei, nrm, hw, out, E);
}